// GAT_8718783611331
// MI455X (gfx1250) — compile-verified
//
#include <hip/hip_runtime.h>
#include <math.h>

// GAT forward, fused per-layer single-pass over adj (537 MB HBM floor ~= 23us).
// LDS-staged, double-buffered attention pass using gfx1250 async global->LDS
// copies (ASYNCcnt) for both the adj stream and the shared h tile, with
// bank-conflict-free LDS strides; WMMA f32 16x16x4 for x@W1 and att@h (out1@W2 is scalar).

typedef __attribute__((ext_vector_type(2))) float v2f;
typedef __attribute__((ext_vector_type(8))) float v8f;
typedef __attribute__((ext_vector_type(4))) int   v4i;

#define GAT_EPS 1e-10f

// ---------------- async global -> LDS plumbing (guarded) --------------------
#if defined(__has_builtin)
#if __has_builtin(__builtin_amdgcn_global_load_async_to_lds_b128)
#define HAVE_ASYNC_LDS 1
#endif
#endif

__device__ __forceinline__ void cp16_async(const float* g, float* l) {
#ifdef HAVE_ASYNC_LDS
    __builtin_amdgcn_global_load_async_to_lds_b128(
        (__attribute__((address_space(1))) v4i*)g,
        (__attribute__((address_space(3))) v4i*)l, 0, 0);
#else
    *(float4*)l = *(const float4*)g;    // synchronous fallback
#endif
}

#ifdef HAVE_ASYNC_LDS
#if __has_builtin(__builtin_amdgcn_s_wait_asynccnt)
#define WAIT_ASYNC(n) __builtin_amdgcn_s_wait_asynccnt(n)
#else
#define WAIT_ASYNC(n) asm volatile("s_wait_asynccnt %0" ::"n"(n) : "memory")
#endif
#else
#define WAIT_ASYNC(n) ((void)0)
#endif

__device__ __forceinline__ float sigmoidf_(float x) {
    return 1.0f / (1.0f + __expf(-x));
}

// ---------------------------------------------------------------------------
// Kernel 1: h1 = x @ W1   (M=8192, K=500, F=64), fp32 WMMA 16x16x4.
// One wave per 16-row M tile; 4 N-tiles of 16 (F=64).
// ---------------------------------------------------------------------------
__global__ void k_gemm_h1(const float* __restrict__ x, const float* __restrict__ W,
                          float* __restrict__ h, int M, int K) {
    const int lane = threadIdx.x & 31;
    const int wid  = (blockIdx.x * blockDim.x + threadIdx.x) >> 5;
    const int row0 = wid * 16;
    if (row0 >= M) return;
    const int half = lane >> 4;
    const int l15  = lane & 15;
    const int r    = row0 + l15;

    v8f c0 = {}, c1 = {}, c2 = {}, c3 = {};
    for (int k0 = 0; k0 < K; k0 += 4) {
        const int kb = k0 + 2 * half;
        v2f a;
        a.x = x[(size_t)r * K + kb];
        a.y = x[(size_t)r * K + kb + 1];
        {
            v2f b; const int n = l15;
            b.x = W[(size_t)kb * 64 + n]; b.y = W[(size_t)(kb + 1) * 64 + n];
            c0 = __builtin_amdgcn_wmma_f32_16x16x4_f32(false, a, false, b, (short)0, c0, false, false);
        }
        {
            v2f b; const int n = 16 + l15;
            b.x = W[(size_t)kb * 64 + n]; b.y = W[(size_t)(kb + 1) * 64 + n];
            c1 = __builtin_amdgcn_wmma_f32_16x16x4_f32(false, a, false, b, (short)0, c1, false, false);
        }
        {
            v2f b; const int n = 32 + l15;
            b.x = W[(size_t)kb * 64 + n]; b.y = W[(size_t)(kb + 1) * 64 + n];
            c2 = __builtin_amdgcn_wmma_f32_16x16x4_f32(false, a, false, b, (short)0, c2, false, false);
        }
        {
            v2f b; const int n = 48 + l15;
            b.x = W[(size_t)kb * 64 + n]; b.y = W[(size_t)(kb + 1) * 64 + n];
            c3 = __builtin_amdgcn_wmma_f32_16x16x4_f32(false, a, false, b, (short)0, c3, false, false);
        }
    }
#pragma unroll
    for (int v = 0; v < 8; ++v) {
        const int m = row0 + v + 8 * half;
        h[(size_t)m * 64 +  0 + l15] = c0[v];
        h[(size_t)m * 64 + 16 + l15] = c1[v];
        h[(size_t)m * 64 + 32 + l15] = c2[v];
        h[(size_t)m * 64 + 48 + l15] = c3[v];
    }
}

// ---------------------------------------------------------------------------
// Kernel 2: per-row attention scores  s_src[i]=h_i·a[:F], s_dst[i]=h_i·a[F:]
// ---------------------------------------------------------------------------
__global__ void k_scores(const float* __restrict__ h, const float* __restrict__ a,
                         float* __restrict__ s_src, float* __restrict__ s_dst,
                         int M, int F, int stride) {
    int i = blockIdx.x * blockDim.x + threadIdx.x;
    if (i >= M) return;
    float acc0 = 0.f, acc1 = 0.f;
    for (int n = 0; n < F; ++n) {
        float v = h[(size_t)i * stride + n];
        acc0 += v * a[n];
        acc1 += v * a[F + n];
    }
    s_src[i] = acc0;
    s_dst[i] = acc1;
}

// ---------------------------------------------------------------------------
// Kernel 3: fused attention pass, LDS-staged + double-buffered async copies.
//
// Block = 4 waves = 64 rows (one 16-row WMMA M-tile per wave). All waves walk
// the j (K) dimension together in chunks of JC=32, sharing LDS tiles:
//   ab: adj[64 x 32] tile, row stride 36 dwords -> conflict-free b64 A reads
//   hb: h[32 x F] tile, row stride 72 (F=64) / 40 (F=16) -> half-waves on
//       disjoint bank ranges for B reads.
// Both C += att@h (WMMA) and per-row sum(att^2) accumulate in one adj pass;
// epilogue scales by 1/(||row||+eps).  Async issue count per thread is
// uniform, so the per-wave ASYNCcnt wait threshold is a compile-time constant.
// ---------------------------------------------------------------------------
template <int NT>
__global__ __launch_bounds__(128) void k_attn2(
        const float* __restrict__ adj, const float* __restrict__ h,
        const float* __restrict__ ssrc, const float* __restrict__ sdst,
        float* __restrict__ out, int N) {
    constexpr int F     = NT * 16;
    constexpr int JC    = 32;                  // j-chunk length
    constexpr int RB    = 64;                  // rows per block
    constexpr int ASTR  = 36;                  // LDS dword stride, adj tile rows
    constexpr int HSTR  = (NT == 4) ? 72 : 40; // LDS dword stride, h tile rows
    constexpr int AG    = (RB * JC / 4) / 128; // adj b128 granules / thread (4)
    constexpr int HG    = (JC * F / 4) / 128;  // h granules / thread (4 or 1)
    constexpr int PER   = AG + HG;             // async ops per wave per chunk

    __shared__ float ab[2][RB * ASTR];
    __shared__ float hb[2][JC * HSTR];

    const int tid  = threadIdx.x;
    const int lane = tid & 31;
    const int wl   = tid >> 5;              // wave index in block
    const int half = lane >> 4;
    const int l15  = lane & 15;
    const int rowB = blockIdx.x * RB;
    const int row0 = rowB + wl * 16;
    const int r    = row0 + l15;

    const float sr = ssrc[r];

    v8f c[NT];
#pragma unroll
    for (int nt = 0; nt < NT; ++nt) c[nt] = (v8f){};
    float ss = 0.f;

    const int NC = N / JC;

    // ---- prologue: stage chunk 0 into buffer 0 ----
#pragma unroll
    for (int k = 0; k < AG; ++k) {
        int gi = tid + k * 128;
        int rr = gi >> 3, c4 = (gi & 7) * 4;           // JC/4 = 8 granules/row
        cp16_async(&adj[(size_t)(rowB + rr) * N + c4], &ab[0][rr * ASTR + c4]);
    }
#pragma unroll
    for (int k = 0; k < HG; ++k) {
        int gi = tid + k * 128;
        int rr = gi / (F / 4), c4 = (gi % (F / 4)) * 4;
        cp16_async(&h[(size_t)rr * F + c4], &hb[0][rr * HSTR + c4]);
    }

    for (int cidx = 0; cidx < NC; ++cidx) {
        const int buf = cidx & 1;
        const int j0  = cidx * JC;

        if (cidx + 1 < NC) {
            const int jn = j0 + JC;
            const int nb = buf ^ 1;
#pragma unroll
            for (int k = 0; k < AG; ++k) {
                int gi = tid + k * 128;
                int rr = gi >> 3, c4 = (gi & 7) * 4;
                cp16_async(&adj[(size_t)(rowB + rr) * N + jn + c4],
                           &ab[nb][rr * ASTR + c4]);
            }
#pragma unroll
            for (int k = 0; k < HG; ++k) {
                int gi = tid + k * 128;
                int rr = gi / (F / 4), c4 = (gi % (F / 4)) * 4;
                cp16_async(&h[(size_t)(jn + rr) * F + c4],
                           &hb[nb][rr * HSTR + c4]);
            }
            WAIT_ASYNC(PER);   // oldest chunk's copies complete (in-order)
        } else {
            WAIT_ASYNC(0);
        }
        __syncthreads();

        const float* abw = &ab[buf][(wl * 16 + l15) * ASTR];
        const float* hbw = &hb[buf][0];
#pragma unroll
        for (int jj = 0; jj < JC; jj += 4) {
            const int jb = jj + 2 * half;
            const float2 av = *(const float2*)&abw[jb];      // adj[r][j0+jb..+1]
            const float sd0 = sdst[j0 + jb];
            const float sd1 = sdst[j0 + jb + 1];
            const float a0 = sigmoidf_(sr + sd0) * av.x;
            const float a1 = sigmoidf_(sr + sd1) * av.y;
            ss += a0 * a0 + a1 * a1;
            v2f a; a.x = a0; a.y = a1;
#pragma unroll
            for (int nt = 0; nt < NT; ++nt) {
                const int n = nt * 16 + l15;
                v2f b;
                b.x = hbw[jb * HSTR + n];
                b.y = hbw[(jb + 1) * HSTR + n];
                c[nt] = __builtin_amdgcn_wmma_f32_16x16x4_f32(false, a, false, b, (short)0, c[nt], false, false);
            }
        }
        __syncthreads();   // all waves done reading before buffer is re-staged
    }

    // Row m's sumsq lives split across lanes m and m+16.
    ss += __shfl_xor(ss, 16, 32);
#pragma unroll
    for (int v = 0; v < 8; ++v) {
        const int m = v + 8 * half;
        const float rn = 1.0f / (sqrtf(__shfl(ss, m, 32)) + GAT_EPS);
#pragma unroll
        for (int nt = 0; nt < NT; ++nt) {
            out[(size_t)(row0 + m) * F + nt * 16 + l15] = c[nt][v] * rn;
        }
    }
}

// ---------------------------------------------------------------------------
// Kernel 4: h2 = out1 @ W2  (8192x64 * 64x10), stored padded to stride 16
// with zeros in cols 10..15 so layer-2 WMMA pass can use one 16-wide N tile.
// ---------------------------------------------------------------------------
__global__ void k_h2(const float* __restrict__ hin, const float* __restrict__ W2,
                     float* __restrict__ h2p, int M) {
    int i = blockIdx.x * blockDim.x + threadIdx.x;
    if (i >= M) return;
    float acc[10];
#pragma unroll
    for (int cc = 0; cc < 10; ++cc) acc[cc] = 0.f;
    for (int k = 0; k < 64; ++k) {
        float v = hin[(size_t)i * 64 + k];
#pragma unroll
        for (int cc = 0; cc < 10; ++cc) acc[cc] += v * W2[k * 10 + cc];
    }
#pragma unroll
    for (int cc = 0; cc < 10; ++cc) h2p[(size_t)i * 16 + cc] = acc[cc];
#pragma unroll
    for (int cc = 10; cc < 16; ++cc) h2p[(size_t)i * 16 + cc] = 0.f;
}

// ---------------------------------------------------------------------------
// Kernel 5: log_softmax over 10 classes, reading padded stride-16 input.
// ---------------------------------------------------------------------------
__global__ void k_logsoftmax(const float* __restrict__ inp, float* __restrict__ out, int M) {
    int i = blockIdx.x * blockDim.x + threadIdx.x;
    if (i >= M) return;
    float v[10];
    float mx = -INFINITY;
#pragma unroll
    for (int cc = 0; cc < 10; ++cc) {
        v[cc] = inp[(size_t)i * 16 + cc];
        mx = fmaxf(mx, v[cc]);
    }
    float s = 0.f;
#pragma unroll
    for (int cc = 0; cc < 10; ++cc) s += __expf(v[cc] - mx);
    const float lse = mx + __logf(s);
#pragma unroll
    for (int cc = 0; cc < 10; ++cc) out[(size_t)i * 10 + cc] = v[cc] - lse;
}

// ---------------------------------------------------------------------------
extern "C" void kernel_launch(void* const* d_in, const int* in_sizes, int n_in,
                              void* d_out, int out_size, void* d_ws, size_t ws_size,
                              hipStream_t stream) {
    const float* x   = (const float*)d_in[0];   // [8192, 500]
    const float* adj = (const float*)d_in[1];   // [8192, 8192]
    const float* W1  = (const float*)d_in[2];   // [500, 64]
    const float* a1  = (const float*)d_in[3];   // [128]
    const float* W2  = (const float*)d_in[4];   // [64, 10]
    const float* a2  = (const float*)d_in[5];   // [20]
    float* out = (float*)d_out;                 // [8192, 10]

    const int N = 8192;
    const int K = in_sizes[0] / N;              // 500

    float* ws  = (float*)d_ws;
    float* h1  = ws;                    // N*64
    float* o1  = h1  + (size_t)N * 64;  // N*64
    float* h2p = o1  + (size_t)N * 64;  // N*16 (padded)
    float* o2p = h2p + (size_t)N * 16;  // N*16 (padded)
    float* s1s = o2p + (size_t)N * 16;  // N
    float* s1d = s1s + N;               // N
    float* s2s = s1d + N;               // N
    float* s2d = s2s + N;               // N

    // ---- Layer 1 ----
    k_gemm_h1<<<128, 128, 0, stream>>>(x, W1, h1, N, K);
    k_scores<<<N / 256, 256, 0, stream>>>(h1, a1, s1s, s1d, N, 64, 64);
    k_attn2<4><<<N / 64, 128, 0, stream>>>(adj, h1, s1s, s1d, o1, N);

    // ---- Layer 2 ----
    k_h2<<<N / 256, 256, 0, stream>>>(o1, W2, h2p, N);
    k_scores<<<N / 256, 256, 0, stream>>>(h2p, a2, s2s, s2d, N, 10, 16);
    k_attn2<1><<<N / 64, 128, 0, stream>>>(adj, h2p, s2s, s2d, o2p, N);

    // ---- Output ----
    k_logsoftmax<<<N / 256, 256, 0, stream>>>(o2p, out, N);
}